// LineGraphic2d_63548336112205
// MI455X (gfx1250) — compile-verified
//
#include <hip/hip_runtime.h>
#include <hip/hip_bf16.h>

// LineGraphic2d on 8192x8192 canvas, fp32 out.
// Roofline: 256 MB stores @ 23.3 TB/s ~= 11 us floor; scalar fma body is
// ~12 VALU issues/px (sqrt co-executes as trans) so compute sits below it.
// Store path: float4/lane -> LDS slot -> GLOBAL_STORE_ASYNC_FROM_LDS_B128 (NT),
// double buffered on ASYNCcnt; per-step global delta lives in the imm offset.

#define CANVAS_H 8192
#define CANVAS_W 8192

typedef int v4i __attribute__((ext_vector_type(4)));
typedef __attribute__((address_space(1))) v4i* gptr_b128;  // global, 16B granule
typedef __attribute__((address_space(3))) v4i* lptr_b128;  // LDS, 16B granule

__device__ __forceinline__ gptr_b128 as_global_b128(const void* p) {
    return (gptr_b128)(unsigned long long)p;
}
__device__ __forceinline__ lptr_b128 as_lds_b128(const void* p) {
    // generic shared pointer: low 32 bits == LDS byte offset (aperture mapping)
    return (lptr_b128)(unsigned)(unsigned long long)p;
}

__global__ __launch_bounds__(256)
void LineGraphic2d_kernel(const float* __restrict__ kp, float* __restrict__ out) {
    __shared__ float4 buf[2][256];   // 8 KB double buffer, per-lane private slots

    const int tid = threadIdx.x;
    const int y   = blockIdx.x;

    // Uniform segment constants (4 floats; uniform address -> s_load_b128)
    const float p0y = kp[0] * (float)CANVAS_H;
    const float p0x = kp[1] * (float)CANVAS_W;
    const float dy  = kp[2] * (float)CANVAS_H - p0y;
    const float dx  = kp[3] * (float)CANVAS_W - p0x;
    const float inv_len2 = 1.0f / (dy * dy + dx * dx);
    const float inv_maxd = 1.0f / 11585.237502960395f;  // 1/(8192*sqrt(2))

    const float ry    = (float)y - p0y;
    const float ry_dy = ry * dy;

    // Single int->float convert for the lane's x base; each step's start x is
    // rx0 + 1024*IT (rx0 carries -p0x, so it need not be an integer and the add may round).
    const float rx0 = (float)(tid * 4) - p0x;

    // One lane base address for the whole row; per-step delta lives in the
    // async store's 24-bit immediate offset (IT * 4096 bytes).
    const float* gbase = out + (size_t)y * CANVAS_W + tid * 4;

#define LG2D_STEP(IT)                                                         \
    {                                                                         \
        /* Reuse guard: at most the previous async store in flight, so   */   \
        /* buffer ((IT) & 1) is free to overwrite.                       */   \
        asm volatile("s_wait_asynccnt 0x1" ::: "memory");                     \
        float rx = rx0 + (float)((IT) * 1024);       /* one v_add per step */ \
        float res[4];                                                         \
        _Pragma("unroll")                                                     \
        for (int j = 0; j < 4; ++j) {                                         \
            float t = fmaf(rx, dx, ry_dy) * inv_len2;         /* fma+mul  */  \
            t = __builtin_amdgcn_fmed3f(t, 0.0f, 1.0f);       /* v_med3   */  \
            const float ey   = fmaf(-t, dy, ry);              /* fma      */  \
            const float ex   = fmaf(-t, dx, rx);              /* fma      */  \
            const float d2   = fmaf(ey, ey, ex * ex);         /* mul+fma  */  \
            const float dist = sqrtf(d2);                     /* trans    */  \
            const float val  = fmaf(dist, -inv_maxd, 0.999f); /* fma      */  \
            res[j] = (dist < 3.0f) ? val : 0.0f;              /* cmp+sel  */  \
            rx += 1.0f;                                       /* add      */  \
        }                                                                     \
        float4* slot = &buf[(IT) & 1][tid];                                   \
        *slot = make_float4(res[0], res[1], res[2], res[3]); /* ds_store */   \
        /* Lane's own LDS write must land before the async engine reads. */   \
        asm volatile("s_wait_dscnt 0x0" ::: "memory");                        \
        /* 16 B/lane LDS -> global, fire-and-forget (ASYNCcnt).          */   \
        /* cpol = 1 (TH_NT): 256 MB written once, keep 192 MB L2 clean.  */   \
        __builtin_amdgcn_global_store_async_from_lds_b128(                    \
            as_global_b128(gbase), as_lds_b128(slot),                         \
            /*offset=*/(IT) * 4096, /*cpol=*/1);                              \
    }

    LG2D_STEP(0) LG2D_STEP(1) LG2D_STEP(2) LG2D_STEP(3)
    LG2D_STEP(4) LG2D_STEP(5) LG2D_STEP(6) LG2D_STEP(7)
#undef LG2D_STEP

    // Drain before wave teardown.
    asm volatile("s_wait_asynccnt 0x0" ::: "memory");
}

extern "C" void kernel_launch(void* const* d_in, const int* in_sizes, int n_in,
                              void* d_out, int out_size, void* d_ws, size_t ws_size,
                              hipStream_t stream) {
    const float* kp  = (const float*)d_in[0];  // key_points: [[y0,x0],[y1,x1]] in [0,1]
    float*       out = (float*)d_out;          // 8192*8192 fp32
    (void)in_sizes; (void)n_in; (void)out_size; (void)d_ws; (void)ws_size;

    LineGraphic2d_kernel<<<dim3(CANVAS_H), dim3(256), 0, stream>>>(kp, out);
}